// Decoder_22703197127089
// MI455X (gfx1250) — compile-verified
//
#include <hip/hip_runtime.h>
#include <math.h>

// ---------------------------------------------------------------------------
// Decoder tail-token pipeline for MI455X (gfx1250), wave32 + f32 WMMA.
//
// Algebraic reduction: no residual connections and only h[:, -1, :] feeds the
// head, so attention/MoE/LN2/head run for the last token only. Heavy work:
//   * conv1d as GEMM  (M=B*N=32736, K=C*P=1536, N=D=64)  ~6.4 GFLOP
//   * K/V projections (M=32736, K=64, N=64 each)          ~0.54 GFLOP
// Both use V_WMMA_F32_16X16X4_F32 (bit-accurate fp32 vs the fp32 reference).
// X traffic = 100 MB -> ~4.3 us at 23.3 TB/s; conv_w (393 KB) is L2-resident.
// Each wave processes a 32-row M-tile so B fragments are amortized over two
// WMMA tiles: conv = 8 WMMA / 6 loads, KV = 16 WMMA / 10 loads per K-step.
// ---------------------------------------------------------------------------

typedef __attribute__((ext_vector_type(2))) float v2f;
typedef __attribute__((ext_vector_type(8))) float v8f;

#define B_    32
#define C_    64
#define L_    12288
#define D_    64
#define E_    8
#define P_    24
#define S_    12
#define PRED_ 96
#define N_    1023                 // (L-P)/S + 1
#define MTOT  (B_ * N_)            // 32736 = 1023 * 32
#define KCONV (C_ * P_)            // 1536

static_assert(MTOT % 32 == 0, "M tiles must be exact");

__device__ inline float waveReduceSum(float v) {
#pragma unroll
  for (int off = 16; off > 0; off >>= 1) v += __shfl_xor(v, off, 32);
  return v;
}

__device__ inline v8f wmma4(v2f a, v2f b, v8f c) {
  // D = A(16x4,f32) * B(4x16,f32) + C(16x16,f32)
  return __builtin_amdgcn_wmma_f32_16x16x4_f32(false, a, false, b, (short)0, c,
                                               false, false);
}

// ---------------------------------------------------------------------------
// Kernel 1: conv1d as WMMA GEMM, fused conv bias + sinusoidal pos-encoding.
// One wave per 32-token M-tile (two 16-row A fragments share the B fragments);
// the wave covers all 4 d-tiles (D=64).
// A[m][kk] = X[b, c, n*S + p]  (kk = c*P + p; a K-group of 4 never crosses c
// since P=24 is a multiple of 4).  B[kk][d'] = conv_w[(dbase+d')*KCONV + kk].
// ---------------------------------------------------------------------------
__global__ __launch_bounds__(32) void conv_wmma_kernel(
    const float* __restrict__ X, const float* __restrict__ W,
    const float* __restrict__ cbias, float* __restrict__ hraw) {
  const int lane = threadIdx.x;
  const int row  = lane & 15;   // A: M-row / B: N-col
  const int kh   = lane >> 4;   // K half (0 -> K=k0,k0+1 ; 1 -> K=k0+2,k0+3)
  const int m0   = (int)blockIdx.x * 32;

  const int mA = m0 + row;
  const int mB = m0 + 16 + row;
  const int bA = mA / N_, nA = mA - bA * N_;
  const int bB = mB / N_, nB = mB - bB * N_;

  const float* xbase0 = X + (size_t)bA * C_ * L_ + nA * S_ + 2 * kh;
  const float* xbase1 = X + (size_t)bB * C_ * L_ + nB * S_ + 2 * kh;
  const float* wrow   = W + (size_t)row * KCONV + 2 * kh;

  v8f acc[2][4] = {};

  int c = 0, p0 = 0;
  for (int k0 = 0; k0 < KCONV; k0 += 4) {
    const size_t xoff = (size_t)c * L_ + p0;
    v2f a0 = *(const v2f*)(xbase0 + xoff);
    v2f a1 = *(const v2f*)(xbase1 + xoff);
    const float* wk = wrow + k0;
    v2f b0 = *(const v2f*)(wk);
    v2f b1 = *(const v2f*)(wk + 16 * KCONV);
    v2f b2 = *(const v2f*)(wk + 32 * KCONV);
    v2f b3 = *(const v2f*)(wk + 48 * KCONV);
    acc[0][0] = wmma4(a0, b0, acc[0][0]);
    acc[0][1] = wmma4(a0, b1, acc[0][1]);
    acc[0][2] = wmma4(a0, b2, acc[0][2]);
    acc[0][3] = wmma4(a0, b3, acc[0][3]);
    acc[1][0] = wmma4(a1, b0, acc[1][0]);
    acc[1][1] = wmma4(a1, b1, acc[1][1]);
    acc[1][2] = wmma4(a1, b2, acc[1][2]);
    acc[1][3] = wmma4(a1, b3, acc[1][3]);
    p0 += 4;
    if (p0 == P_) { p0 = 0; ++c; }
  }

  const float NLOG1E4_D = -9.210340371976184f / (float)D_;  // -ln(1e4)/D
#pragma unroll
  for (int dt = 0; dt < 4; ++dt) {
    const int d    = dt * 16 + row;
    const float f  = __expf(NLOG1E4_D * (float)(2 * (d >> 1)));
    const float cb = cbias[d];
#pragma unroll
    for (int half = 0; half < 2; ++half) {
#pragma unroll
      for (int r = 0; r < 8; ++r) {
        const int mm = m0 + half * 16 + r + 8 * kh;  // C/D row mapping
        const int bb = mm / N_;
        const int nn = mm - bb * N_;
        const float pe = (d & 1) ? cosf(nn * f) : sinf(nn * f);
        hraw[(size_t)mm * D_ + d] = acc[half][dt][r] + cb + pe;
      }
    }
  }
}

// ---------------------------------------------------------------------------
// Kernel 2: LayerNorm over D=64, one wave32 per token (2 elems per lane).
// ---------------------------------------------------------------------------
__global__ __launch_bounds__(256) void ln_kernel(
    const float* __restrict__ x, const float* __restrict__ g,
    const float* __restrict__ bta, float* __restrict__ y) {
  const int t    = (int)blockIdx.x * 8 + (threadIdx.x >> 5);
  const int lane = threadIdx.x & 31;
  const float* xr = x + (size_t)t * D_;
  const float x0 = xr[lane], x1 = xr[lane + 32];
  const float mean = waveReduceSum(x0 + x1) * (1.0f / 64.0f);
  const float d0 = x0 - mean, d1 = x1 - mean;
  const float var = waveReduceSum(d0 * d0 + d1 * d1) * (1.0f / 64.0f);
  const float inv = rsqrtf(var + 1e-5f);
  float* yr = y + (size_t)t * D_;
  yr[lane]      = d0 * inv * g[lane] + bta[lane];
  yr[lane + 32] = d1 * inv * g[lane + 32] + bta[lane + 32];
}

// ---------------------------------------------------------------------------
// Kernel 3: fused K and V projections via WMMA (M=32736, K=64, N=64).
// One wave per 32-row M-tile; B fragments (kw/vw) shared across both halves.
// ---------------------------------------------------------------------------
__global__ __launch_bounds__(32) void kv_wmma_kernel(
    const float* __restrict__ h, const float* __restrict__ kw,
    const float* __restrict__ kb, const float* __restrict__ vw,
    const float* __restrict__ vb, float* __restrict__ Km,
    float* __restrict__ Vm) {
  const int lane = threadIdx.x;
  const int row  = lane & 15;
  const int kh   = lane >> 4;
  const int m0   = (int)blockIdx.x * 32;

  const float* arow0 = h  + (size_t)(m0 + row) * D_ + 2 * kh;
  const float* arow1 = h  + (size_t)(m0 + 16 + row) * D_ + 2 * kh;
  const float* krow  = kw + (size_t)row * D_ + 2 * kh;
  const float* vrow  = vw + (size_t)row * D_ + 2 * kh;

  v8f accK[2][4] = {};
  v8f accV[2][4] = {};

#pragma unroll 4
  for (int k0 = 0; k0 < D_; k0 += 4) {
    v2f a0 = *(const v2f*)(arow0 + k0);
    v2f a1 = *(const v2f*)(arow1 + k0);
#pragma unroll
    for (int dt = 0; dt < 4; ++dt) {
      v2f bk = *(const v2f*)(krow + dt * 16 * D_ + k0);
      v2f bv = *(const v2f*)(vrow + dt * 16 * D_ + k0);
      accK[0][dt] = wmma4(a0, bk, accK[0][dt]);
      accK[1][dt] = wmma4(a1, bk, accK[1][dt]);
      accV[0][dt] = wmma4(a0, bv, accV[0][dt]);
      accV[1][dt] = wmma4(a1, bv, accV[1][dt]);
    }
  }

#pragma unroll
  for (int dt = 0; dt < 4; ++dt) {
    const int d = dt * 16 + row;
    const float bk = kb[d], bv = vb[d];
#pragma unroll
    for (int half = 0; half < 2; ++half) {
#pragma unroll
      for (int r = 0; r < 8; ++r) {
        const int mm = m0 + half * 16 + r + 8 * kh;
        Km[(size_t)mm * D_ + d] = accK[half][dt][r] + bk;
        Vm[(size_t)mm * D_ + d] = accV[half][dt][r] + bv;
      }
    }
  }
}

// ---------------------------------------------------------------------------
// Kernel 4: everything that only the last token needs. One block per batch b.
// q-last, one softmax row over N=1023, V-sum, o-proj, router softmax + top-4,
// 4 selected experts, moe-out proj, LN2, 96-way head.
// ---------------------------------------------------------------------------
__global__ __launch_bounds__(256) void tail_kernel(
    const float* __restrict__ h, const float* __restrict__ Km,
    const float* __restrict__ Vm, const float* __restrict__ qw,
    const float* __restrict__ qb, const float* __restrict__ ow,
    const float* __restrict__ ob, const float* __restrict__ ew,
    const float* __restrict__ eb, const float* __restrict__ rw,
    const float* __restrict__ rb, const float* __restrict__ mw,
    const float* __restrict__ mb, const float* __restrict__ g2,
    const float* __restrict__ b2, const float* __restrict__ outw,
    const float* __restrict__ outb, float* __restrict__ out) {
  __shared__ float sh[64];        // h[b, N-1, :]
  __shared__ float sq[64];        // q last token
  __shared__ float sscore[N_];    // softmax row
  __shared__ float red[256];
  __shared__ float part[4][64];
  __shared__ float sh1[64];       // after o-proj (MoE input)
  __shared__ float smoe[64];
  __shared__ float sh2[64];       // after moe_out proj
  __shared__ float sy[64];        // after LN2
  __shared__ float slog[E_];
  __shared__ int   stopi[4];
  __shared__ float stopw[4];
  __shared__ float stat[2];

  const int b   = blockIdx.x;
  const int tid = threadIdx.x;
  const size_t base = (size_t)b * N_ * D_;

  if (tid < 64) sh[tid] = h[base + (size_t)(N_ - 1) * D_ + tid];
  __syncthreads();

  // q = h_last @ qw.T + qb
  if (tid < 64) {
    float s = qb[tid];
    const float* wr = qw + tid * D_;
    for (int j = 0; j < D_; ++j) s += sh[j] * wr[j];
    sq[tid] = s;
  }
  __syncthreads();

  // scores[m] = q . K[b,m,:] / 8   (causal mask is all-valid on the last row;
  // the reference's uniform +1.0 cancels in softmax)
  for (int m = tid; m < N_; m += 256) {
    const float* kr = Km + base + (size_t)m * D_;
    float s = 0.0f;
    for (int j = 0; j < D_; ++j) s += sq[j] * kr[j];
    sscore[m] = s * 0.125f;
  }
  __syncthreads();

  // softmax: block max
  float lm = -1e30f;
  for (int m = tid; m < N_; m += 256) lm = fmaxf(lm, sscore[m]);
  red[tid] = lm;
  __syncthreads();
  for (int off = 128; off > 0; off >>= 1) {
    if (tid < off) red[tid] = fmaxf(red[tid], red[tid + off]);
    __syncthreads();
  }
  const float smax = red[0];
  __syncthreads();
  float ls = 0.0f;
  for (int m = tid; m < N_; m += 256) {
    const float e = __expf(sscore[m] - smax);
    sscore[m] = e;
    ls += e;
  }
  red[tid] = ls;
  __syncthreads();
  for (int off = 128; off > 0; off >>= 1) {
    if (tid < off) red[tid] += red[tid + off];
    __syncthreads();
  }
  const float invsum = 1.0f / red[0];
  __syncthreads();

  // attn[d] = sum_m w[m] * V[b,m,d]
  {
    const int grp = tid >> 6, d = tid & 63;
    float acc = 0.0f;
    for (int m = grp; m < N_; m += 4)
      acc += sscore[m] * Vm[base + (size_t)m * D_ + d];
    part[grp][d] = acc;
  }
  __syncthreads();
  if (tid < 64)
    part[0][tid] =
        (part[0][tid] + part[1][tid] + part[2][tid] + part[3][tid]) * invsum;
  __syncthreads();

  // o-projection
  if (tid < 64) {
    float s = ob[tid];
    const float* wr = ow + tid * D_;
    for (int j = 0; j < D_; ++j) s += part[0][j] * wr[j];
    sh1[tid] = s;
  }
  __syncthreads();

  // router logits + softmax + top-4
  if (tid < E_) {
    float s = rb[tid];
    const float* wr = rw + tid * D_;
    for (int j = 0; j < D_; ++j) s += sh1[j] * wr[j];
    slog[tid] = s;
  }
  __syncthreads();
  if (tid == 0) {
    float mx = slog[0];
    for (int e = 1; e < E_; ++e) mx = fmaxf(mx, slog[e]);
    float ex[E_], ssum = 0.0f;
    for (int e = 0; e < E_; ++e) { ex[e] = __expf(slog[e] - mx); ssum += ex[e]; }
    for (int e = 0; e < E_; ++e) ex[e] /= ssum;
    bool used[E_] = {};
    for (int k = 0; k < 4; ++k) {
      int best = 0; float bv = -1.0f;
      for (int e = 0; e < E_; ++e)
        if (!used[e] && ex[e] > bv) { bv = ex[e]; best = e; }
      used[best] = true;
      stopi[k] = best;
      stopw[k] = bv;
    }
  }
  __syncthreads();

  // 4 selected experts, weighted
  {
    const int kk = tid >> 6, o = tid & 63;
    const int e = stopi[kk];
    float s = eb[e * D_ + o];
    const float* wr = ew + (size_t)e * D_ * D_ + o * D_;
    for (int j = 0; j < D_; ++j) s += sh1[j] * wr[j];
    part[kk][o] = stopw[kk] * s;
  }
  __syncthreads();
  if (tid < 64)
    smoe[tid] = part[0][tid] + part[1][tid] + part[2][tid] + part[3][tid];
  __syncthreads();

  // moe_out projection
  if (tid < 64) {
    float s = mb[tid];
    const float* wr = mw + tid * D_;
    for (int j = 0; j < D_; ++j) s += smoe[j] * wr[j];
    sh2[tid] = s;
  }
  __syncthreads();

  // LN2
  if (tid == 0) {
    float mn = 0.0f;
    for (int j = 0; j < D_; ++j) mn += sh2[j];
    mn *= (1.0f / 64.0f);
    float vv = 0.0f;
    for (int j = 0; j < D_; ++j) { const float d = sh2[j] - mn; vv += d * d; }
    vv *= (1.0f / 64.0f);
    stat[0] = mn;
    stat[1] = rsqrtf(vv + 1e-5f);
  }
  __syncthreads();
  if (tid < 64) sy[tid] = (sh2[tid] - stat[0]) * stat[1] * g2[tid] + b2[tid];
  __syncthreads();

  // head: (PRED=96) x (D=64)
  if (tid < PRED_) {
    float s = outb[tid];
    const float* wr = outw + tid * D_;
    for (int j = 0; j < D_; ++j) s += sy[j] * wr[j];
    out[b * PRED_ + tid] = s;
  }
}

// ---------------------------------------------------------------------------
extern "C" void kernel_launch(void* const* d_in, const int* in_sizes, int n_in,
                              void* d_out, int out_size, void* d_ws,
                              size_t ws_size, hipStream_t stream) {
  (void)in_sizes; (void)n_in; (void)out_size; (void)ws_size;
  const float* X    = (const float*)d_in[0];
  const float* cw   = (const float*)d_in[1];
  const float* cb   = (const float*)d_in[2];
  const float* g1   = (const float*)d_in[3];
  const float* b1   = (const float*)d_in[4];
  const float* qw   = (const float*)d_in[5];
  const float* qb   = (const float*)d_in[6];
  const float* kw   = (const float*)d_in[7];
  const float* kb   = (const float*)d_in[8];
  const float* vw   = (const float*)d_in[9];
  const float* vb   = (const float*)d_in[10];
  const float* ow   = (const float*)d_in[11];
  const float* ob   = (const float*)d_in[12];
  const float* ew   = (const float*)d_in[13];
  const float* eb   = (const float*)d_in[14];
  const float* rw   = (const float*)d_in[15];
  const float* rb   = (const float*)d_in[16];
  const float* mw   = (const float*)d_in[17];
  const float* mb   = (const float*)d_in[18];
  const float* g2   = (const float*)d_in[19];
  const float* b2   = (const float*)d_in[20];
  const float* outw = (const float*)d_in[21];
  const float* outb = (const float*)d_in[22];
  float* out = (float*)d_out;

  float* ws = (float*)d_ws;
  const size_t MN = (size_t)MTOT * D_;   // 2,095,104 floats
  float* hraw = ws;                      // pre-LN conv output
  float* hn   = ws + MN;                 // LN1 output
  float* Km   = ws + 2 * MN;             // K projection
  float* Vm   = ws + 3 * MN;             // V projection  (total ~33.5 MB)

  conv_wmma_kernel<<<MTOT / 32, 32, 0, stream>>>(X, cw, cb, hraw);
  ln_kernel<<<MTOT / 8, 256, 0, stream>>>(hraw, g1, b1, hn);
  kv_wmma_kernel<<<MTOT / 32, 32, 0, stream>>>(hn, kw, kb, vw, vb, Km, Vm);
  tail_kernel<<<B_, 256, 0, stream>>>(hn, Km, Vm, qw, qb, ow, ob, ew, eb, rw,
                                      rb, mw, mb, g2, b2, outw, outb, out);
}